// StructInductionNoSplit_43026982371925
// MI455X (gfx1250) — compile-verified
//
#include <hip/hip_runtime.h>

typedef __attribute__((ext_vector_type(2))) float v2f;
typedef __attribute__((ext_vector_type(8))) float v8f;

#define Bsz 32
#define Tsz 512
#define Dsz 256

// ---------------- workspace layout (float offsets) ----------------
// aug  : [B][512][1024]  (64 MB)  -- Q (16 MB) and K (16 MB) overlay its start
// A    : [B][512][512]   (32 MB)
// cs   : [B][512] column sums
// fi   : [B][512] root scores
#define AUG_OFF 0
#define Q_OFF   0
#define K_OFF   4194304
#define A_OFF   16777216
#define CS_OFF  25165824
#define FI_OFF  (25165824 + 16384)

// ---------------- tiny init: zero column sums ----------------
__global__ void si_zero_cs(float* __restrict__ cs) {
    int i = blockIdx.x * blockDim.x + threadIdx.x;
    if (i < Bsz * Tsz) cs[i] = 0.0f;
}

// ---------------- f_i = x . wr + br   (one wave per row) ----------------
__global__ void si_fi(const float* __restrict__ x, const float* __restrict__ wr,
                      const float* __restrict__ br, float* __restrict__ fi) {
    const int lane = threadIdx.x & 31;
    const int wave = threadIdx.x >> 5;
    const int row  = blockIdx.x * (blockDim.x >> 5) + wave;   // 0..B*T-1
    const float* xr = x + (size_t)row * Dsz;
    float s = 0.0f;
    for (int k = lane; k < Dsz; k += 32) s += xr[k] * wr[k];
    for (int off = 16; off > 0; off >>= 1) s += __shfl_down(s, off, 32);
    if (lane == 0) fi[row] = s + br[0];
}

// ---------------- fused K/Q projection with V_WMMA_F32_16X16X4_F32 ----------------
// grid (1024, 2), block 128 (4 waves). Wave computes a 16x32 strip of K AND Q
// (4 accumulators) so the A fragment is loaded once per k-step and feeds 4 WMMAs.
__global__ void si_proj(const float* __restrict__ x,
                        const float* __restrict__ Wk, const float* __restrict__ bk,
                        const float* __restrict__ Wq, const float* __restrict__ bq,
                        float* __restrict__ Qo, float* __restrict__ Ko) {
    const int lane  = threadIdx.x & 31;
    const int wave  = threadIdx.x >> 5;
    const int rt    = blockIdx.x;                        // row tile: 0..1023
    const int ct0   = (blockIdx.y * 4 + wave) * 2;       // col tiles ct0, ct0+1 (0..15)
    const int row   = rt * 16 + (lane & 15);
    const int col0  = ct0 * 16 + (lane & 15);
    const int col1  = col0 + 16;
    const int khalf = (lane >> 4) * 2;                   // lanes 16-31 hold K+2,K+3

    v8f accK0 = {}, accK1 = {}, accQ0 = {}, accQ1 = {};
    const float* xr = x + (size_t)row * Dsz;
    for (int kk = 0; kk < Dsz; kk += 4) {
        const int k = kk + khalf;
        v2f a = *(const v2f*)(xr + k);                   // A frag, shared by 4 WMMAs
        v2f bK0, bK1, bQ0, bQ1;
        bK0.x = Wk[(size_t)k * Dsz + col0]; bK0.y = Wk[(size_t)(k + 1) * Dsz + col0];
        bK1.x = Wk[(size_t)k * Dsz + col1]; bK1.y = Wk[(size_t)(k + 1) * Dsz + col1];
        bQ0.x = Wq[(size_t)k * Dsz + col0]; bQ0.y = Wq[(size_t)(k + 1) * Dsz + col0];
        bQ1.x = Wq[(size_t)k * Dsz + col1]; bQ1.y = Wq[(size_t)(k + 1) * Dsz + col1];
        accK0 = __builtin_amdgcn_wmma_f32_16x16x4_f32(false, a, false, bK0, (short)0, accK0, false, false);
        accK1 = __builtin_amdgcn_wmma_f32_16x16x4_f32(false, a, false, bK1, (short)0, accK1, false, false);
        accQ0 = __builtin_amdgcn_wmma_f32_16x16x4_f32(false, a, false, bQ0, (short)0, accQ0, false, false);
        accQ1 = __builtin_amdgcn_wmma_f32_16x16x4_f32(false, a, false, bQ1, (short)0, accQ1, false, false);
    }
    const float bk0 = bk[col0], bk1 = bk[col1];
    const float bq0 = bq[col0], bq1 = bq[col1];
    const int mhi = (lane >> 4) * 8;
    #pragma unroll
    for (int v = 0; v < 8; ++v) {
        const int r = rt * 16 + v + mhi;
        Ko[(size_t)r * Dsz + col0] = accK0[v] + bk0;
        Ko[(size_t)r * Dsz + col1] = accK1[v] + bk1;
        Qo[(size_t)r * Dsz + col0] = (accQ0[v] + bq0) * 0.0625f;   // 1/sqrt(256)
        Qo[(size_t)r * Dsz + col1] = (accQ1[v] + bq1) * 0.0625f;
    }
}

// ---------------- A = exp(Q K^T) (diag=0) + column sums ----------------
// grid (B, 32 i-tiles, 2), block 128: wave handles a 16x64 strip (4 j-tiles),
// so the Q fragment is loaded once per k-step and feeds 4 WMMAs.
__global__ void si_attn(const float* __restrict__ Q, const float* __restrict__ K,
                        float* __restrict__ A, float* __restrict__ cs) {
    const int b    = blockIdx.x;
    const int it   = blockIdx.y;
    const int lane = threadIdx.x & 31;
    const int wave = threadIdx.x >> 5;
    const int jt0  = (blockIdx.z * 4 + wave) * 4;        // j tiles jt0..jt0+3
    const int khalf = (lane >> 4) * 2;

    const float* Qb = Q + (size_t)b * Tsz * Dsz;
    const float* Kb = K + (size_t)b * Tsz * Dsz;
    const int qi = it * 16 + (lane & 15);
    const float* qr  = Qb + (size_t)qi * Dsz;
    const float* kr0 = Kb + (size_t)((jt0 + 0) * 16 + (lane & 15)) * Dsz;
    const float* kr1 = Kb + (size_t)((jt0 + 1) * 16 + (lane & 15)) * Dsz;
    const float* kr2 = Kb + (size_t)((jt0 + 2) * 16 + (lane & 15)) * Dsz;
    const float* kr3 = Kb + (size_t)((jt0 + 3) * 16 + (lane & 15)) * Dsz;

    v8f acc0 = {}, acc1 = {}, acc2 = {}, acc3 = {};
    for (int kk = 0; kk < Dsz; kk += 4) {
        const int k = kk + khalf;
        v2f a  = *(const v2f*)(qr + k);                  // A frag, shared
        v2f b0 = *(const v2f*)(kr0 + k);                 // B[k][n] = key[n][k]
        v2f b1 = *(const v2f*)(kr1 + k);
        v2f b2 = *(const v2f*)(kr2 + k);
        v2f b3 = *(const v2f*)(kr3 + k);
        acc0 = __builtin_amdgcn_wmma_f32_16x16x4_f32(false, a, false, b0, (short)0, acc0, false, false);
        acc1 = __builtin_amdgcn_wmma_f32_16x16x4_f32(false, a, false, b1, (short)0, acc1, false, false);
        acc2 = __builtin_amdgcn_wmma_f32_16x16x4_f32(false, a, false, b2, (short)0, acc2, false, false);
        acc3 = __builtin_amdgcn_wmma_f32_16x16x4_f32(false, a, false, b3, (short)0, acc3, false, false);
    }
    float* Ab = A + (size_t)b * Tsz * Tsz;
    const int mhi = (lane >> 4) * 8;
    v8f accs[4] = {acc0, acc1, acc2, acc3};
    #pragma unroll
    for (int t = 0; t < 4; ++t) {
        const int j = (jt0 + t) * 16 + (lane & 15);
        float csum = 0.0f;
        #pragma unroll
        for (int v = 0; v < 8; ++v) {
            const int i = it * 16 + v + mhi;
            const float e = (i == j) ? 0.0f : __expf(accs[t][v]);
            Ab[(size_t)i * Tsz + j] = e;
            csum += e;
        }
        atomicAdd(&cs[b * Tsz + j], csum);
    }
}

// ---------------- build augmented [L_bar | I] ----------------
__global__ void si_build(const float* __restrict__ A, const float* __restrict__ cs,
                         const float* __restrict__ fi, float* __restrict__ aug) {
    const long long idx = (long long)blockIdx.x * blockDim.x + threadIdx.x;
    if (idx >= (long long)Bsz * Tsz * 1024) return;
    const int b = (int)(idx >> 19);            // 512*1024 = 2^19
    const int rem = (int)(idx & ((1 << 19) - 1));
    const int i = rem >> 10;
    const int j = rem & 1023;
    float val;
    if (j < Tsz) {
        if (i == 0) val = fi[b * Tsz + j];
        else        val = ((i == j) ? cs[b * Tsz + j] : 0.0f) - A[(size_t)b * Tsz * Tsz + (size_t)i * Tsz + j];
    } else {
        val = (i == (j - Tsz)) ? 1.0f : 0.0f;
    }
    aug[idx] = val;
}

// ---------------- Gauss-Jordan inverse, one workgroup per batch ----------------
// aug[b]: 512 x 1024.  After the loop the right half holds L_bar^{-1}.
__global__ void si_gj(float* __restrict__ aug) {
    __shared__ float lf[Tsz];
    float* m = aug + (size_t)blockIdx.x * Tsz * 1024;
    const int tid = threadIdx.x;               // 0..1023 = column index
    for (int p = 0; p < Tsz; ++p) {
        __syncthreads();
        const float pivot = m[(size_t)p * 1024 + p];
        const float ipiv  = 1.0f / pivot;
        const float prow  = m[(size_t)p * 1024 + tid] * ipiv;
        if (tid < Tsz) lf[tid] = m[(size_t)tid * 1024 + p];   // old column p
        m[(size_t)p * 1024 + tid] = prow;                     // scaled pivot row
        __syncthreads();
        if (p + 1 < Tsz)                                       // hide next pivot-row latency
            __builtin_prefetch(&m[(size_t)(p + 1) * 1024 + tid], 0, 3);
        #pragma unroll 4
        for (int r = 0; r < Tsz; ++r) {
            if (r == p) continue;
            m[(size_t)r * 1024 + tid] -= lf[r] * prow;
        }
    }
}

// ---------------- assemble df[B][T][T+1] ----------------
__global__ void si_out(const float* __restrict__ A, const float* __restrict__ fi,
                       const float* __restrict__ aug, float* __restrict__ out) {
    const long long idx = (long long)blockIdx.x * blockDim.x + threadIdx.x;
    const long long total = (long long)Bsz * Tsz * (Tsz + 1);
    if (idx >= total) return;
    const int b   = (int)(idx / ((long long)Tsz * (Tsz + 1)));
    const int rem = (int)(idx % ((long long)Tsz * (Tsz + 1)));
    const int j = rem / (Tsz + 1);
    const int k = rem % (Tsz + 1);
    const float* inv = aug + (size_t)b * Tsz * 1024 + Tsz;    // inv(r,c) = inv[r*1024+c]
    float val;
    if (k == 0) {
        val = fi[b * Tsz + j] * inv[(size_t)j * 1024 + 0];    // d0
    } else {
        const int i = k - 1;
        const float a = A[(size_t)b * Tsz * Tsz + (size_t)i * Tsz + j];
        const float t1 = (j != 0) ? a * inv[(size_t)j * 1024 + j] : 0.0f;  // A_ij * inv_jj
        const float t2 = (i != 0) ? a * inv[(size_t)j * 1024 + i] : 0.0f;  // A_ij * inv_ji
        val = t1 - t2;
    }
    out[idx] = val;
}

extern "C" void kernel_launch(void* const* d_in, const int* in_sizes, int n_in,
                              void* d_out, int out_size, void* d_ws, size_t ws_size,
                              hipStream_t stream) {
    const float* x  = (const float*)d_in[0];
    const float* Wk = (const float*)d_in[1];
    const float* bk = (const float*)d_in[2];
    const float* Wq = (const float*)d_in[3];
    const float* bq = (const float*)d_in[4];
    const float* wr = (const float*)d_in[5];
    const float* br = (const float*)d_in[6];
    float* ws  = (float*)d_ws;
    float* aug = ws + AUG_OFF;
    float* Qo  = ws + Q_OFF;     // overlays aug (consumed before si_build writes)
    float* Ko  = ws + K_OFF;
    float* A   = ws + A_OFF;
    float* cs  = ws + CS_OFF;
    float* fi  = ws + FI_OFF;
    float* out = (float*)d_out;

    si_zero_cs<<<(Bsz * Tsz + 255) / 256, 256, 0, stream>>>(cs);
    si_fi<<<(Bsz * Tsz) / 8, 256, 0, stream>>>(x, wr, br, fi);
    si_proj<<<dim3((Bsz * Tsz) / 16, 2), 128, 0, stream>>>(x, Wk, bk, Wq, bq, Qo, Ko);
    si_attn<<<dim3(Bsz, Tsz / 16, 2), 128, 0, stream>>>(Qo, Ko, A, cs);
    {
        long long total = (long long)Bsz * Tsz * 1024;
        si_build<<<(unsigned)((total + 255) / 256), 256, 0, stream>>>(A, cs, fi, aug);
    }
    si_gj<<<Bsz, 1024, 0, stream>>>(aug);
    {
        long long total = (long long)Bsz * Tsz * (Tsz + 1);
        si_out<<<(unsigned)((total + 255) / 256), 256, 0, stream>>>(A, fi, aug, out);
    }
}